// DepthLossForImgBEV_56255481643354
// MI455X (gfx1250) — compile-verified
//
#include <hip/hip_runtime.h>

// ---------------------------------------------------------------------------
// DepthLossForImgBEV: weighted BCE over (B, N, D, H, W), reduced to a scalar.
//
// Memory-bound: ~60.6 MB of depth logits streamed once -> ~2.6 us floor at
// 23.3 TB/s. Structure: 1 thread per GT pixel (B*N*H*W = 135168), each thread
// loops the D=112 depth bins (per-iteration accesses are coalesced across the
// wave: consecutive lanes hit consecutive hw). Invalid pixels (gt==0, ~20%)
// skip the whole loop -> 20% less DRAM traffic.
//
// Reduction: per-wave via V_WMMA_F32_16X16X4_F32 with an all-ones B matrix
// (rowsum trick), then LDS across waves, per-block partial to d_ws, and a
// deterministic single-block finish kernel (no float atomics -> bitwise
// reproducible across graph replays).
// ---------------------------------------------------------------------------

typedef float v2f __attribute__((ext_vector_type(2)));
typedef float v8f __attribute__((ext_vector_type(8)));

#define HW_PIX 11264   // H*W = 64*176 (fixed by reference setup_inputs)
#define DBINS  112     // len(arange(2.0, 58.0, 0.5))

// Sum of all 32 lanes' values, returned in every lane.
// A(16x4): lanes 0-15 VGPR0 -> A[m][0], lanes 16-31 VGPR0 -> A[m][2]; VGPR1=0.
// B all-ones => D[m][n] = rowsum(m) = x[m] + x[m+16]. Summing the 8 D VGPRs
// gives sum(rows 0-7) in lanes 0-15 and sum(rows 8-15) in lanes 16-31; one
// xor-16 shuffle completes the 32-lane total.
__device__ __forceinline__ float wave_sum32_wmma(float x) {
    v2f a; a[0] = x;    a[1] = 0.0f;
    v2f b; b[0] = 1.0f; b[1] = 1.0f;
    v8f c = {};
    v8f d = __builtin_amdgcn_wmma_f32_16x16x4_f32(
        /*neg_a=*/false, a, /*neg_b=*/false, b,
        /*c_mod=*/(short)0, c, /*reuse_a=*/false, /*reuse_b=*/false);
    float s = ((d[0] + d[1]) + (d[2] + d[3])) + ((d[4] + d[5]) + (d[6] + d[7]));
    s += __shfl_xor(s, 16, 32);   // combine the two lane halves (wave32)
    return s;
}

__device__ __forceinline__ float block_sum(float v, float* lds8) {
    float ws = wave_sum32_wmma(v);
    int lane = threadIdx.x & 31;
    int wave = threadIdx.x >> 5;
    if (lane == 0) lds8[wave] = ws;
    __syncthreads();
    float s = 0.0f;
    if (threadIdx.x == 0) {
        int nw = (blockDim.x + 31) >> 5;
        for (int i = 0; i < nw; ++i) s += lds8[i];
    }
    return s;   // valid in thread 0 only
}

__global__ void depth_loss_partial_kernel(const float* __restrict__ depth_gt,
                                          const float* __restrict__ depth,
                                          float* __restrict__ partials,
                                          int npix) {
    int p = blockIdx.x * blockDim.x + threadIdx.x;
    float acc = 0.0f;
    if (p < npix) {
        float g = depth_gt[p];
        if (g != 0.0f) {                       // weight==0 pixels contribute 0
            // idx = clip(floor((gt - 2.0)/0.5), 0, 112)
            float f   = floorf((g - 2.0f) * 2.0f);
            int   idx = (int)fminf(fmaxf(f, 0.0f), 112.0f);
            int hw = p % HW_PIX;
            int bn = p / HW_PIX;               // flattened (b*N + n)
            const float* xp = depth + (size_t)bn * DBINS * HW_PIX + hw;
            #pragma unroll 4
            for (int d = 0; d < DBINS; ++d) {
                float x  = xp[(size_t)d * HW_PIX];
                // log_sigmoid(x)  = min(x,0) - log1p(exp(-|x|))
                // log_sigmoid(-x) = -max(x,0) - log1p(exp(-|x|))
                float z      = __logf(1.0f + __expf(-fabsf(x)));
                float log_p  = fmaxf(fminf(x, 0.0f) - z, -100.0f);
                float log_1m = fmaxf(-fmaxf(x, 0.0f) - z, -100.0f);
                // target==1 iff d == idx (idx==112 never matches -> zero row)
                acc -= (d == idx) ? log_p : log_1m;
            }
        }
    }
    __shared__ float lds8[8];
    float s = block_sum(acc, lds8);
    if (threadIdx.x == 0) partials[blockIdx.x] = s;
}

__global__ void depth_loss_final_kernel(const float* __restrict__ partials,
                                        int nparts,
                                        float* __restrict__ out,
                                        float scale) {
    float acc = 0.0f;
    for (int i = threadIdx.x; i < nparts; i += blockDim.x) acc += partials[i];
    __shared__ float lds8[8];
    float s = block_sum(acc, lds8);
    if (threadIdx.x == 0) out[0] = s * scale;   // scale = 3.0 / numel
}

extern "C" void kernel_launch(void* const* d_in, const int* in_sizes, int n_in,
                              void* d_out, int out_size, void* d_ws, size_t ws_size,
                              hipStream_t stream) {
    const float* depth_gt = (const float*)d_in[0];  // (B,N,H,W) f32
    const float* depth    = (const float*)d_in[1];  // (B,N*D,H,W) f32
    float* out            = (float*)d_out;          // scalar f32
    float* partials       = (float*)d_ws;           // one f32 per block

    int npix   = in_sizes[0];                       // B*N*H*W = 135168
    long long  ntot = (long long)in_sizes[1];       // B*N*D*H*W = 15138816

    const int threads = 256;                        // 8 waves / block (wave32)
    int blocks = (npix + threads - 1) / threads;    // 528

    depth_loss_partial_kernel<<<blocks, threads, 0, stream>>>(
        depth_gt, depth, partials, npix);
    depth_loss_final_kernel<<<1, threads, 0, stream>>>(
        partials, blocks, out, 3.0f / (float)ntot);
}